// WeightedPooling_90967407329694
// MI455X (gfx1250) — compile-verified
//
#include <hip/hip_runtime.h>

typedef float v2f __attribute__((ext_vector_type(2)));
typedef float v8f __attribute__((ext_vector_type(8)));

#define MDIM 64
#define LDM  68          // padded LDS leading dim: bank-conflict-free for both A and B access
#define NTHREADS 512     // 16 wave32 waves -> 16 x (16x16) tiles = full 64x64 product

// ---------------------------------------------------------------------------
// 64x64x64 f32 matmul on the WMMA pipe. Each of the 16 waves computes one
// 16x16 tile of D = A*B with 16 chained V_WMMA_F32_16X16X4_F32 ops.
// A,B are LDS matrices with leading dimension LDM. Result stays in VGPRs
// (v8f accumulator, C/D layout per ISA 7.12.2) so callers can do register-
// staged in-place updates with explicit barriers.
// ---------------------------------------------------------------------------
__device__ __forceinline__ v8f mmul64(const float* A, const float* B, int tid) {
  const int wave = tid >> 5;
  const int lane = tid & 31;
  const int tm   = wave >> 2;     // tile row 0..3
  const int tn   = wave & 3;      // tile col 0..3
  const int half = lane >> 4;     // 0 -> K{0,1}, 1 -> K{2,3}
  const int m    = lane & 15;
  const float* arow = A + (tm * 16 + m) * LDM + half * 2;   // A[M=m][k..]
  const float* bcol = B + (tn * 16 + m) + half * 2 * LDM;   // B[k..][N=m]
  v8f acc = {};
#pragma unroll
  for (int k0 = 0; k0 < MDIM; k0 += 4) {
    v2f a, b;
    a.x = arow[k0];
    a.y = arow[k0 + 1];
    b.x = bcol[k0 * LDM];
    b.y = bcol[k0 * LDM + LDM];
    acc = __builtin_amdgcn_wmma_f32_16x16x4_f32(false, a, false, b,
                                                (short)0, acc, false, false);
  }
  return acc;
}

__device__ __forceinline__ void tile_coords(int tid, int& r0, int& col) {
  const int wave = tid >> 5;
  const int lane = tid & 31;
  r0  = (wave >> 2) * 16 + (lane >> 4) * 8;   // VGPR v holds rows r0+v
  col = (wave & 3) * 16 + (lane & 15);
}

__device__ __forceinline__ void tile_store(float* D, v8f acc, int tid) {
  int r0, col; tile_coords(tid, r0, col);
#pragma unroll
  for (int v = 0; v < 8; ++v) D[(r0 + v) * LDM + col] = acc[v];
}

// D = 1.5*I - 0.5*acc   (Newton-Schulz helper matrix)
__device__ __forceinline__ void tile_store_ns(float* D, v8f acc, int tid) {
  int r0, col; tile_coords(tid, r0, col);
#pragma unroll
  for (int v = 0; v < 8; ++v) {
    const int row = r0 + v;
    D[row * LDM + col] = ((row == col) ? 1.5f : 0.0f) - 0.5f * acc[v];
  }
}

// D = I + p*acc   (Horner step for expm)
__device__ __forceinline__ void tile_store_axpyI(float* D, v8f acc, float p, int tid) {
  int r0, col; tile_coords(tid, r0, col);
#pragma unroll
  for (int v = 0; v < 8; ++v) {
    const int row = r0 + v;
    D[row * LDM + col] = ((row == col) ? 1.0f : 0.0f) + p * acc[v];
  }
}

// ---------------------------------------------------------------------------
// Row softmax of the 64x64 weight matrix (tiny).
// ---------------------------------------------------------------------------
__global__ void __launch_bounds__(64)
softmax_rows(const float* __restrict__ W, float* __restrict__ Wsm) {
  __shared__ float red[64];
  const int i = blockIdx.x;
  const int t = threadIdx.x;
  const float v = W[i * 64 + t];
  red[t] = v; __syncthreads();
  for (int o = 32; o > 0; o >>= 1) {
    if (t < o) red[t] = fmaxf(red[t], red[t + o]);
    __syncthreads();
  }
  const float mx = red[0]; __syncthreads();
  const float e = expf(v - mx);
  red[t] = e; __syncthreads();
  for (int o = 32; o > 0; o >>= 1) {
    if (t < o) red[t] += red[t + o];
    __syncthreads();
  }
  Wsm[i * 64 + t] = e / red[0];
}

// ---------------------------------------------------------------------------
// Phase 1: L = logm(X) for one SPD 64x64 matrix per block.
// Inverse scaling-and-squaring: X/s (s = ||X||_F), 5 Newton-Schulz matrix
// square roots, then 8-term Taylor log; logX = 32*log(X^(1/32)/..) + ln(s)*I.
// ---------------------------------------------------------------------------
__global__ void __launch_bounds__(NTHREADS)
logm_kernel(const float* __restrict__ X, float* __restrict__ Lout) {
  __shared__ float y[MDIM * LDM];   // M / Y / R / E
  __shared__ float z[MDIM * LDM];   // Z / power buffer
  __shared__ float t[MDIM * LDM];   // NS helper
  __shared__ float red[NTHREADS];
  const int tid = threadIdx.x;
  const int g   = blockIdx.x;
  const int row = tid >> 3;          // each thread owns 8 contiguous elems
  const int c0  = (tid & 7) * 8;

  const float* Xg = X + (size_t)g * (MDIM * MDIM) + row * MDIM + c0;
  const float4 x0 = ((const float4*)Xg)[0];
  const float4 x1 = ((const float4*)Xg)[1];
  float xs[8] = {x0.x, x0.y, x0.z, x0.w, x1.x, x1.y, x1.z, x1.w};

  // Frobenius norm (deterministic tree reduction)
  float ss = 0.f;
#pragma unroll
  for (int j = 0; j < 8; ++j) ss += xs[j] * xs[j];
  red[tid] = ss; __syncthreads();
  for (int o = NTHREADS / 2; o > 0; o >>= 1) {
    if (tid < o) red[tid] += red[tid + o];
    __syncthreads();
  }
  const float s    = sqrtf(red[0]);
  const float invs = 1.0f / s;

  // y = X / s  (spectrum in (0,1])
#pragma unroll
  for (int j = 0; j < 8; ++j) y[row * LDM + c0 + j] = xs[j] * invs;

  // 5 repeated sqrt via coupled Newton-Schulz; later roots converge faster
#pragma unroll 1
  for (int step = 0; step < 5; ++step) {
    const int iters = (step == 0) ? 14 : (step == 1) ? 9 : (step == 2) ? 6
                    : (step == 3) ? 4 : 3;
#pragma unroll
    for (int j = 0; j < 8; ++j)
      z[row * LDM + c0 + j] = (row == c0 + j) ? 1.0f : 0.0f;   // Z = I
    __syncthreads();
#pragma unroll 1
    for (int it = 0; it < iters; ++it) {
      v8f a = mmul64(z, y, tid);        // Z*Y
      tile_store_ns(t, a, tid);         // T = 1.5I - 0.5*Z*Y
      __syncthreads();
      v8f ay = mmul64(y, t, tid);       // Ynew = Y*T
      __syncthreads();                  // all reads of y done
      tile_store(y, ay, tid);
      __syncthreads();
      v8f az = mmul64(t, z, tid);       // Znew = T*Z
      __syncthreads();                  // all reads of z done
      tile_store(z, az, tid);
      __syncthreads();
    }
  }

  // E = R - I (in place in y); series accumulator lives in registers
  float acc8[8];
#pragma unroll
  for (int j = 0; j < 8; ++j) {
    const float e = y[row * LDM + c0 + j] - ((row == c0 + j) ? 1.0f : 0.0f);
    y[row * LDM + c0 + j] = e;
    z[row * LDM + c0 + j] = e;          // P1 = E
    acc8[j] = e;                        // term 1 coefficient = +1
  }
  __syncthreads();

  // log(I+E) = E - E^2/2 + E^3/3 - ... (terms 2..8)
#pragma unroll 1
  for (int term = 2; term <= 8; ++term) {
    v8f q = mmul64(z, y, tid);          // P <- P*E
    __syncthreads();                    // all reads of z done
    tile_store(z, q, tid);
    __syncthreads();
    const float cf = ((term & 1) ? 1.0f : -1.0f) / (float)term;
#pragma unroll
    for (int j = 0; j < 8; ++j) acc8[j] += cf * z[row * LDM + c0 + j];
  }

  const float lgs = logf(s);
  float* Lg = Lout + (size_t)g * (MDIM * MDIM) + row * MDIM + c0;
#pragma unroll
  for (int j = 0; j < 8; ++j)
    Lg[j] = 32.0f * acc8[j] + ((row == c0 + j) ? lgs : 0.0f);
}

// ---------------------------------------------------------------------------
// Phase 3: per (b,i) block: S = sum_c W[i,c] * L[b,c]  (L2-resident reads),
// then expm(S) by scaling-and-squaring with order-8 Horner Taylor.
// ---------------------------------------------------------------------------
__global__ void __launch_bounds__(NTHREADS)
mixexp_kernel(const float* __restrict__ Lws, const float* __restrict__ Wsm,
              float* __restrict__ out) {
  __shared__ float Tm[MDIM * LDM];
  __shared__ float Am[MDIM * LDM];
  __shared__ float wrow[MDIM];
  __shared__ float red[NTHREADS];
  const int tid = threadIdx.x;
  const int g   = blockIdx.x;
  const int b   = g >> 6;
  const int i   = g & 63;
  if (tid < MDIM) wrow[tid] = Wsm[i * MDIM + tid];
  __syncthreads();

  const int row = tid >> 3;
  const int c0  = (tid & 7) * 8;
  const float* Lb = Lws + (size_t)b * (64 * MDIM * MDIM) + row * MDIM + c0;

  // channel mix (elementwise over c; VALU FMAs, L fits in L2)
  float sacc[8] = {0.f, 0.f, 0.f, 0.f, 0.f, 0.f, 0.f, 0.f};
#pragma unroll 4
  for (int c = 0; c < 64; ++c) {
    const float wc = wrow[c];
    const float4* p4 = (const float4*)(Lb + (size_t)c * (MDIM * MDIM));
    const float4 a0 = p4[0];
    const float4 a1 = p4[1];
    sacc[0] += wc * a0.x; sacc[1] += wc * a0.y;
    sacc[2] += wc * a0.z; sacc[3] += wc * a0.w;
    sacc[4] += wc * a1.x; sacc[5] += wc * a1.y;
    sacc[6] += wc * a1.z; sacc[7] += wc * a1.w;
  }

  // Frobenius norm of S -> number of squarings
  float ss = 0.f;
#pragma unroll
  for (int j = 0; j < 8; ++j) ss += sacc[j] * sacc[j];
  red[tid] = ss; __syncthreads();
  for (int o = NTHREADS / 2; o > 0; o >>= 1) {
    if (tid < o) red[tid] += red[tid + o];
    __syncthreads();
  }
  const float nrm = sqrtf(red[0]);
  int sq = 0;
  float th = nrm;
  while (th > 0.25f && sq < 30) { th *= 0.5f; ++sq; }   // uniform in block
  const float scale = __int_as_float((127 - sq) << 23); // 2^-sq

  // T = S*2^-sq ; A = I + T/8 (highest Horner term)
#pragma unroll
  for (int j = 0; j < 8; ++j) {
    const float tv = sacc[j] * scale;
    Tm[row * LDM + c0 + j] = tv;
    Am[row * LDM + c0 + j] = ((row == c0 + j) ? 1.0f : 0.0f) + tv * 0.125f;
  }
  __syncthreads();

  // Horner: A <- I + (T*A)/k, k = 7..1  (in-place via register staging)
#pragma unroll 1
  for (int k = 7; k >= 1; --k) {
    v8f p = mmul64(Tm, Am, tid);
    __syncthreads();
    tile_store_axpyI(Am, p, 1.0f / (float)k, tid);
    __syncthreads();
  }

  // sq squarings (runtime-uniform loop)
#pragma unroll 1
  for (int q = 0; q < sq; ++q) {
    v8f p = mmul64(Am, Am, tid);
    __syncthreads();
    tile_store(Am, p, tid);
    __syncthreads();
  }

  float* Og = out + (size_t)g * (MDIM * MDIM) + row * MDIM + c0;
#pragma unroll
  for (int j = 0; j < 8; ++j) Og[j] = Am[row * LDM + c0 + j];
}

// ---------------------------------------------------------------------------
// d_in[0]: x [64,64,64,64] f32 ; d_in[1]: weights [64,64] f32
// d_out:  [64,64,64,64] f32
// d_ws:   16 KB softmax(W) + 64 MB matrix-log scratch
// ---------------------------------------------------------------------------
extern "C" void kernel_launch(void* const* d_in, const int* in_sizes, int n_in,
                              void* d_out, int out_size, void* d_ws, size_t ws_size,
                              hipStream_t stream) {
  const float* x = (const float*)d_in[0];
  const float* w = (const float*)d_in[1];
  float* out = (float*)d_out;
  float* Wsm = (float*)d_ws;                 // 64*64 floats
  float* Lws = Wsm + 64 * 64;                // nmat * 4096 floats
  const int nmat = in_sizes[0] / (MDIM * MDIM);   // B*C = 4096
  (void)n_in; (void)out_size; (void)ws_size;

  softmax_rows<<<dim3(64), dim3(64), 0, stream>>>(w, Wsm);
  logm_kernel<<<dim3(nmat), dim3(NTHREADS), 0, stream>>>(x, Lws);
  mixexp_kernel<<<dim3(nmat), dim3(NTHREADS), 0, stream>>>(Lws, Wsm, out);
}